// InferenceEmbeddingTable_19129784336957
// MI455X (gfx1250) — compile-verified
//
#include <hip/hip_runtime.h>
#include <hip/hip_bf16.h>

// ---------------------------------------------------------------------------
// Hash-bucket embedding lookup (InferenceEmbeddingTable), MI455X / gfx1250.
// Pure gather/scatter op (no matmul -> no WMMA). Memory-bound:
// ~522 MB HBM traffic -> ~22us floor @ 23.3 TB/s. Strategy:
//   * wave32-per-index bucket probe: 32 lanes x int4 = 512B bucket in one
//     b128 load, resolved with __ballot + v_readlane (CDNA5 wave32 idiom)
//   * 8-deep batches per wave for memory-level parallelism (Little's law)
//   * embedding-row copy via async mem->LDS->mem (global_load_async_to_lds_
//     b128 / global_store_async_from_lds_b128, ASYNCcnt), VGPR-free
//   * aggressive scalarization: readfirstlane(waveid) + readlane broadcasts
//     keep all per-index bookkeeping on the SALU port (uniform values),
//     leaving VALU mostly idle and vmem addresses in SADDR+voffset form
//   * global_prefetch for bucket rows, nontemporal streaming output stores
// ---------------------------------------------------------------------------

typedef float f32x4 __attribute__((ext_vector_type(4)));
typedef int   i32x4 __attribute__((ext_vector_type(4)));

// Async-LDS builtins take int4* pointers in global (AS1) / LDS (AS3) spaces.
typedef __attribute__((address_space(1))) i32x4 gas_i32x4;  // global
typedef __attribute__((address_space(3))) i32x4 las_i32x4;  // LDS

#if __has_builtin(__builtin_amdgcn_global_load_async_to_lds_b128) && \
    __has_builtin(__builtin_amdgcn_global_store_async_from_lds_b128)
#define USE_ASYNC_LDS 1
#else
#define USE_ASYNC_LDS 0
#endif

__device__ __forceinline__ void wait_async_0() {
#if __has_builtin(__builtin_amdgcn_s_wait_asynccnt)
  __builtin_amdgcn_s_wait_asynccnt(0);
#else
  asm volatile("s_wait_asynccnt 0x0" ::: "memory");
#endif
}

#define EMB_DIM    128
#define BUCKET_CAP 128
#define WPB        8   // waves per 256-thread block (wave32)
#define BATCH      8   // rows in flight per wave per phase

__global__ __launch_bounds__(256)
void emb_bucket_lookup_kernel(const int* __restrict__ indices,
                              const int* __restrict__ offsets,
                              const int* __restrict__ bucket_keys,
                              const int* __restrict__ table_offsets,
                              const int* __restrict__ table_bucket_offsets,
                              const int* __restrict__ num_buckets,
                              const float* __restrict__ emb,
                              float* __restrict__ out,
                              int N, int n_tables)
{
#if USE_ASYNC_LDS
  __shared__ float stage[WPB][BATCH][EMB_DIM];  // 32 KB / block staging
#endif
  const int lane = threadIdx.x & 31;
  // Scalarize the wave id: all chunk/base/address bookkeeping goes SALU.
  const int wv = __builtin_amdgcn_readfirstlane(threadIdx.x >> 5);
  const int wavesPerGrid = gridDim.x * (blockDim.x >> 5);
  const int waveId = blockIdx.x * (blockDim.x >> 5) + wv;

  for (long long base = (long long)waveId * 32; base < (long long)N;
       base += (long long)wavesPerGrid * 32) {
    const long long rem = (long long)N - base;
    const int cnt = rem < 32 ? (int)rem : 32;

    // Per-lane precompute for this wave's 32 indices (fully parallel).
    int myIdx = -1, myBG = 0, myRowBase = 0;
    if (lane < cnt) {
      const int gi = (int)base + lane;
      myIdx = indices[gi];
      int tid = 0;  // searchsorted(offsets, gi, 'right') - 1
      while (tid + 1 < n_tables && gi >= offsets[tid + 1]) ++tid;
      const int nb = num_buckets[tid];
      const int bl = ((nb & (nb - 1)) == 0) ? (myIdx & (nb - 1)) : (myIdx % nb);
      myBG = table_bucket_offsets[tid] + bl;
      myRowBase = table_offsets[tid] + bl * BUCKET_CAP;
      // Pull this lane's bucket row toward the WGP (global_prefetch_b8).
      __builtin_prefetch(bucket_keys + (size_t)myBG * BUCKET_CAP, 0, 3);
    }

    for (int b = 0; b < 32; b += BATCH) {
      const int nvalid = (cnt - b) < BATCH ? (cnt - b) : BATCH;
      if (nvalid <= 0) break;

      // Phase 1: issue BATCH bucket-probe loads (512B each, one per index).
      // readlane -> SGPR bucket id: scalar 64-bit base + lane*16 voffset.
      i32x4 kv[BATCH];
      int idxs[BATCH];
#pragma unroll
      for (int u = 0; u < BATCH; ++u) {
        idxs[u] = __builtin_amdgcn_readlane(myIdx, b + u);
        const int bg_u = __builtin_amdgcn_readlane(myBG, b + u);
        if (u < nvalid)
          kv[u] = ((const i32x4*)(bucket_keys + (size_t)bg_u * BUCKET_CAP))[lane];
      }

      // Phase 2: resolve slot via ballot; compute global embedding row
      // entirely in scalar registers (bal/fl/slot/row are wave-uniform).
      long long rows[BATCH];
      unsigned fmask = 0;
#pragma unroll
      for (int u = 0; u < BATCH; ++u) {
        rows[u] = 0;
        if (u < nvalid) {
          const int tgt = idxs[u];
          int m = 4;  // first-match priority (argmax over booleans)
          if (kv[u].w == tgt) m = 3;
          if (kv[u].z == tgt) m = 2;
          if (kv[u].y == tgt) m = 1;
          if (kv[u].x == tgt) m = 0;
          const unsigned long long bal = __ballot(m < 4);
          if (bal != 0ull) {
            const int fl = __ffsll(bal) - 1;
            const int slot = (fl << 2) + __builtin_amdgcn_readlane(m, fl);
            rows[u] =
                (long long)__builtin_amdgcn_readlane(myRowBase, b + u) + slot;
            fmask |= 1u << u;
          }
        }
      }

      // Phase 3: copy rows (512B each).
#if USE_ASYNC_LDS
      wait_async_0();  // WAR: prior async stores out of these LDS buffers
#pragma unroll
      for (int u = 0; u < BATCH; ++u) {
        if (u < nvalid && (fmask & (1u << u))) {
          const float* src = emb + (size_t)rows[u] * EMB_DIM + lane * 4;
          __builtin_amdgcn_global_load_async_to_lds_b128(
              (gas_i32x4*)src, (las_i32x4*)&stage[wv][u][lane * 4], 0, 0);
        }
      }
      wait_async_0();  // RAW: all BATCH rows resident in LDS
#pragma unroll
      for (int u = 0; u < BATCH; ++u) {
        if (u < nvalid) {
          float* dst = out + (size_t)(base + b + u) * EMB_DIM + lane * 4;
          if (fmask & (1u << u)) {
            __builtin_amdgcn_global_store_async_from_lds_b128(
                (gas_i32x4*)dst, (las_i32x4*)&stage[wv][u][lane * 4], 0, 0);
          } else {
            f32x4 z = {0.f, 0.f, 0.f, 0.f};
            __builtin_nontemporal_store(z, (f32x4*)dst);
          }
        }
      }
#else
      // Fallback: direct VGPR b128 copy, 8 loads in flight, NT streaming.
      f32x4 v[BATCH];
#pragma unroll
      for (int u = 0; u < BATCH; ++u) {
        f32x4 z = {0.f, 0.f, 0.f, 0.f};
        v[u] = z;
        if (u < nvalid && (fmask & (1u << u)))
          v[u] = __builtin_nontemporal_load(
              (const f32x4*)(emb + (size_t)rows[u] * EMB_DIM) + lane);
      }
#pragma unroll
      for (int u = 0; u < BATCH; ++u)
        if (u < nvalid)
          __builtin_nontemporal_store(
              v[u], (f32x4*)(out + (size_t)(base + b + u) * EMB_DIM) + lane);
#endif
    }
  }
}

extern "C" void kernel_launch(void* const* d_in, const int* in_sizes, int n_in,
                              void* d_out, int out_size, void* d_ws, size_t ws_size,
                              hipStream_t stream) {
  const int*   indices              = (const int*)d_in[0];
  const int*   offsets              = (const int*)d_in[1];
  const int*   bucket_keys          = (const int*)d_in[2];
  const int*   table_offsets        = (const int*)d_in[3];
  const int*   table_bucket_offsets = (const int*)d_in[4];
  const int*   num_buckets          = (const int*)d_in[5];
  const float* emb                  = (const float*)d_in[6];
  float*       out                  = (float*)d_out;

  const int N = in_sizes[0];
  const int n_tables = in_sizes[1] - 1;

  const int chunks = (N + 31) / 32;            // one 32-index chunk per wave
  int blocks = (chunks + WPB - 1) / WPB;       // 8 waves per block
  if (blocks < 1) blocks = 1;

  emb_bucket_lookup_kernel<<<blocks, 32 * WPB, 0, stream>>>(
      indices, offsets, bucket_keys, table_offsets, table_bucket_offsets,
      num_buckets, emb, out, N, n_tables);
}